// KernelUpdateHead_42039139893425
// MI455X (gfx1250) — compile-verified
//
#include <hip/hip_runtime.h>
#include <hip/hip_bf16.h>
#include <math.h>

// Problem constants (from reference)
#define B_   2
#define N_   100
#define C_   256
#define H_   128
#define W_   128
#define HW_  (H_*W_)      // 16384
#define TOPK 100
#define NH_  8
#define DH_  32
#define G_   32
#define CPG  (C_/G_)      // 8
#define FFN_ 2048
#define EPS_ 1e-5f

typedef float v2f __attribute__((ext_vector_type(2)));
typedef float v8f __attribute__((ext_vector_type(8)));

#define FLAG_RELU 1
#define MT 4              // M tiles (16 rows each) per wave -> 64-row strip

// ---------------------------------------------------------------------------
// Generic fp32 WMMA GEMM: C[z] = act( A[z] (MxK) * B[z] (KxN) + bias + resid )
// One wave (32 threads) per 64(M) x 16(N) strip: the B fragment is loaded once
// per K-step and reused by MT=4 V_WMMA_F32_16X16X4_F32 ops (4x less L2 traffic
// on the streamed operand). M may be ragged (clamped loads keep EXEC all-1s
// across WMMA; stores are guarded). N % 16 == 0 and K % 4 == 0 for all calls.
// ---------------------------------------------------------------------------
__global__ void wmma_gemm_f32(const float* __restrict__ A,  long sA, int lda,
                              const float* __restrict__ Bw, long sB, int ldb,
                              const float* __restrict__ bias,
                              const float* __restrict__ resid, long sR,
                              float* __restrict__ Cm, long sC, int ldc,
                              int M, int Kd, int flags)
{
    const int lane = threadIdx.x & 31;
    const int half = lane >> 4;        // 0: lanes 0-15, 1: lanes 16-31
    const int l16  = lane & 15;
    const int tn = blockIdx.x * 16;
    const int tm = blockIdx.y * (16 * MT);
    const long z = blockIdx.z;

    const float* Ab = A  + z * sA;
    const float* Bb = Bw + z * sB;

    const float* arow[MT];
#pragma unroll
    for (int j = 0; j < MT; ++j) {
        int rowA = tm + 16 * j + l16;      // A-matrix row (16x4 layout)
        if (rowA > M - 1) rowA = M - 1;    // clamp: loads stay valid, EXEC all-1s
        arow[j] = Ab + (long)rowA * lda;
    }

    v8f acc[MT];
#pragma unroll
    for (int j = 0; j < MT; ++j) acc[j] = (v8f){0.f,0.f,0.f,0.f,0.f,0.f,0.f,0.f};

    for (int k0 = 0; k0 < Kd; k0 += 4) {
        const int ka = k0 + 2 * half;
        // B 4x16 f32 fragment (rows ka, ka+1 striped across lane halves):
        // loaded ONCE and fed to all MT WMMAs.
        v2f b;
        b.x = Bb[(long)ka       * ldb + tn + l16];
        b.y = Bb[(long)(ka + 1) * ldb + tn + l16];
        // Prefetch the streamed operand two K-steps ahead (global_prefetch_b8);
        // condition is wave-uniform so EXEC is untouched around the WMMAs.
        if (k0 + 8 < Kd)
            __builtin_prefetch(&Bb[(long)(ka + 8) * ldb + tn + l16], 0, 1);
#pragma unroll
        for (int j = 0; j < MT; ++j) {
            v2f a;
            a.x = arow[j][ka];
            a.y = arow[j][ka + 1];
            acc[j] = __builtin_amdgcn_wmma_f32_16x16x4_f32(
                         false, a, false, b, (short)0, acc[j], false, false);
        }
    }

    const int n = tn + l16;
    const float bv = bias ? bias[n] : 0.f;
#pragma unroll
    for (int j = 0; j < MT; ++j) {
#pragma unroll
        for (int r = 0; r < 8; ++r) {
            const int m = tm + 16 * j + r + 8 * half;  // C layout: VGPR r -> rows r, r+8
            if (m < M) {
                float v = acc[j][r] + bv;
                if (resid) v += resid[z * sR + (long)m * ldc + n];
                if (flags & FLAG_RELU) v = fmaxf(v, 0.f);
                Cm[z * sC + (long)m * ldc + n] = v;
            }
        }
    }
}

// ---------------------------------------------------------------------------
// GroupNorm statistics: one block per (b,g). Channels of a group are
// contiguous in bchw, so each block reduces a contiguous 8*HW span.
// ---------------------------------------------------------------------------
__global__ void gn_stats_kernel(const float* __restrict__ y, float* __restrict__ stats)
{
    const int b = blockIdx.x / G_;
    const int g = blockIdx.x % G_;
    const float* base = y + ((long)b * C_ + (long)g * CPG) * HW_;
    const long total = (long)CPG * HW_;

    float s = 0.f, ss = 0.f;
    for (long i = threadIdx.x; i < total; i += blockDim.x) {
        float v = base[i];
        s += v; ss += v * v;
    }
    __shared__ float r0[256], r1[256];
    const int tid = threadIdx.x;
    r0[tid] = s; r1[tid] = ss;
    __syncthreads();
    for (int off = 128; off; off >>= 1) {
        if (tid < off) { r0[tid] += r0[tid + off]; r1[tid] += r1[tid + off]; }
        __syncthreads();
    }
    if (tid == 0) {
        const float mu  = r0[0] / (float)total;
        const float var = r1[0] / (float)total - mu * mu;
        stats[blockIdx.x * 2]     = mu;
        stats[blockIdx.x * 2 + 1] = rsqrtf(var + EPS_);
    }
}

// In-place GN scale/shift + ReLU (pure elementwise)
__global__ void gn_apply_kernel(float* __restrict__ y, const float* __restrict__ stats,
                                const float* __restrict__ gg, const float* __restrict__ gb)
{
    const long idx = (long)blockIdx.x * blockDim.x + threadIdx.x;  // < B*C*HW (pow2)
    const int c = (int)((idx >> 14) & 255);     // HW = 2^14
    const int b = (int)(idx >> 22);             // C*HW = 2^22
    const int g = c >> 3;                       // 8 channels per group
    const float mu = stats[(b * G_ + g) * 2];
    const float rs = stats[(b * G_ + g) * 2 + 1];
    const float v  = (y[idx] - mu) * rs * gg[c] + gb[c];
    y[idx] = fmaxf(v, 0.f);
}

// ---------------------------------------------------------------------------
// Top-K=100 of 16384 via 4-pass radix select on order-preserving uint keys.
// Order of emitted indices is irrelevant (softmax over K is permutation
// invariant). One block per (b,n).
// ---------------------------------------------------------------------------
__device__ __forceinline__ unsigned fkey(float x)
{
    unsigned u = __float_as_uint(x);
    return (u & 0x80000000u) ? ~u : (u | 0x80000000u);
}

__global__ void topk_kernel(const float* __restrict__ preds, int* __restrict__ inds)
{
    const int bn = blockIdx.x;
    const float* row = preds + (long)bn * HW_;
    const int tid = threadIdx.x;

    __shared__ int hist[256];
    __shared__ unsigned sh_prefix;
    __shared__ int sh_need, sh_high, sh_tie;

    if (tid == 0) { sh_prefix = 0u; sh_need = TOPK; }
    __syncthreads();

    for (int pass = 3; pass >= 0; --pass) {
        for (int i = tid; i < 256; i += blockDim.x) hist[i] = 0;
        __syncthreads();
        const unsigned pmask = (pass == 3) ? 0u : (0xFFFFFFFFu << ((pass + 1) * 8));
        const unsigned pref  = sh_prefix;
        for (int i = tid; i < HW_; i += blockDim.x) {
            const unsigned key = fkey(row[i]);
            if ((key & pmask) == (pref & pmask))
                atomicAdd(&hist[(key >> (pass * 8)) & 255], 1);
        }
        __syncthreads();
        if (tid == 0) {
            int need = sh_need, accum = 0, sel = 0;
            for (int v = 255; v >= 0; --v) {
                if (accum + hist[v] >= need) { sel = v; break; }
                accum += hist[v];
            }
            sh_prefix = pref | ((unsigned)sel << (pass * 8));
            sh_need   = need - accum;
        }
        __syncthreads();
    }

    const unsigned T  = sh_prefix;
    const int needTie = sh_need;
    const int numHigh = TOPK - needTie;
    if (tid == 0) { sh_high = 0; sh_tie = 0; }
    __syncthreads();

    for (int i = tid; i < HW_; i += blockDim.x) {
        const unsigned key = fkey(row[i]);
        if (key > T) {
            const int p = atomicAdd(&sh_high, 1);
            inds[bn * TOPK + p] = i;
        } else if (key == T) {
            const int t = atomicAdd(&sh_tie, 1);
            if (t < needTie) inds[bn * TOPK + numHigh + t] = i;
        }
    }
}

// Gather: feats[bn,k,c] = y[b,c,idx] (coalesced writes, L2-resident reads)
__global__ void gather_kernel(const float* __restrict__ y, const int* __restrict__ inds,
                              float* __restrict__ feats)
{
    const int bnk = blockIdx.x;
    const int k  = bnk % TOPK;
    const int bn = bnk / TOPK;
    const int b  = bn / N_;
    const int idx = inds[bn * TOPK + k];
    const int c = threadIdx.x;
    feats[(long)bnk * C_ + c] = y[((long)b * C_ + c) * HW_ + idx];
}

// ---------------------------------------------------------------------------
// Attention: 1 query x 100 keys x 8 heads. One block per (b,n); one wave32
// per head; wave shuffles for softmax reductions.
// ---------------------------------------------------------------------------
__global__ void attn_kernel(const float* __restrict__ q, const float* __restrict__ kf,
                            const float* __restrict__ vf, float* __restrict__ out)
{
    const int bn   = blockIdx.x;
    const int h    = threadIdx.x >> 5;   // head = wave
    const int lane = threadIdx.x & 31;

    __shared__ float p_lds[NH_][TOPK];

    const float* qh = q + (long)bn * C_ + h * DH_;
    const float scale = 0.1767766952966369f;   // 1/sqrt(32)

    float sc[4];
#pragma unroll
    for (int t = 0; t < 4; ++t) {
        const int k = t * 32 + lane;
        float a = -1e30f;
        if (k < TOPK) {
            const float* kv = kf + ((long)bn * TOPK + k) * C_ + h * DH_;
            float d = 0.f;
            for (int i = 0; i < DH_; ++i) d += qh[i] * kv[i];
            a = d * scale;
        }
        sc[t] = a;
    }
    float mx = fmaxf(fmaxf(sc[0], sc[1]), fmaxf(sc[2], sc[3]));
    for (int off = 16; off; off >>= 1) mx = fmaxf(mx, __shfl_xor(mx, off, 32));

    float sum = 0.f;
#pragma unroll
    for (int t = 0; t < 4; ++t) {
        const int k = t * 32 + lane;
        const float e = (k < TOPK) ? __expf(sc[t] - mx) : 0.f;
        sum += e;
        if (k < TOPK) p_lds[h][k] = e;
    }
    for (int off = 16; off; off >>= 1) sum += __shfl_xor(sum, off, 32);
    __syncthreads();

    const float inv = 1.f / sum;
    const float* vb = vf + (long)bn * TOPK * C_ + h * DH_ + lane;
    float acc = 0.f;
    for (int k = 0; k < TOPK; ++k) acc += p_lds[h][k] * vb[(long)k * C_];
    out[(long)bn * C_ + h * DH_ + lane] = acc * inv;
}

// LayerNorm over C=256: one block per row, one thread per channel.
__global__ void layernorm_kernel(const float* __restrict__ in, const float* __restrict__ g,
                                 const float* __restrict__ bta, float* __restrict__ out,
                                 float* __restrict__ out2, int relu)
{
    const int row = blockIdx.x, c = threadIdx.x;
    const float v = in[(long)row * C_ + c];
    __shared__ float r0[C_], r1[C_];
    __shared__ float smu, srs;
    r0[c] = v; r1[c] = v * v;
    __syncthreads();
    for (int off = 128; off; off >>= 1) {
        if (c < off) { r0[c] += r0[c + off]; r1[c] += r1[c + off]; }
        __syncthreads();
    }
    if (c == 0) {
        const float mu = r0[0] / (float)C_;
        smu = mu;
        srs = rsqrtf(r1[0] / (float)C_ - mu * mu + EPS_);
    }
    __syncthreads();
    float o = (v - smu) * srs * g[c] + bta[c];
    if (relu) o = fmaxf(o, 0.f);
    out[(long)row * C_ + c] = o;
    if (out2) out2[(long)row * C_ + c] = o;
}

// Half-pixel bilinear 2x upsample: (B,N,H,W) -> (B,N,2H,2W)
__global__ void upsample_kernel(const float* __restrict__ lo, float* __restrict__ out)
{
    const long idx = (long)blockIdx.x * blockDim.x + threadIdx.x;
    const int j = (int)(idx % (2 * W_));
    const int i = (int)((idx / (2 * W_)) % (2 * H_));
    const long bn = idx / ((long)4 * HW_);

    float sy = (i + 0.5f) * 0.5f - 0.5f;
    float sx = (j + 0.5f) * 0.5f - 0.5f;
    sy = fminf(fmaxf(sy, 0.f), (float)(H_ - 1));
    sx = fminf(fmaxf(sx, 0.f), (float)(W_ - 1));
    const int y0 = (int)sy, x0 = (int)sx;
    const int y1 = (y0 + 1 < H_) ? y0 + 1 : H_ - 1;
    const int x1 = (x0 + 1 < W_) ? x0 + 1 : W_ - 1;
    const float fy = sy - y0, fx = sx - x0;

    const float* p = lo + bn * HW_;
    const float v00 = p[y0 * W_ + x0], v01 = p[y0 * W_ + x1];
    const float v10 = p[y1 * W_ + x0], v11 = p[y1 * W_ + x1];
    out[idx] = v00 * (1.f - fy) * (1.f - fx) + v01 * (1.f - fy) * fx
             + v10 * fy * (1.f - fx)         + v11 * fy * fx;
}

// ---------------------------------------------------------------------------
extern "C" void kernel_launch(void* const* d_in, const int* in_sizes, int n_in,
                              void* d_out, int out_size, void* d_ws, size_t ws_size,
                              hipStream_t stream)
{
    (void)in_sizes; (void)n_in; (void)out_size; (void)ws_size;

    const float* x      = (const float*)d_in[0];
    const float* pfeat  = (const float*)d_in[1];
    const float* preds  = (const float*)d_in[2];
    const float* conv_w = (const float*)d_in[3];
    const float* gn_g   = (const float*)d_in[4];
    const float* gn_b   = (const float*)d_in[5];
    const float* wq = (const float*)d_in[6];   const float* bq = (const float*)d_in[7];
    const float* wk = (const float*)d_in[8];   const float* bk = (const float*)d_in[9];
    const float* wv = (const float*)d_in[10];  const float* bv = (const float*)d_in[11];
    const float* wo = (const float*)d_in[12];  const float* bo = (const float*)d_in[13];
    const float* tn_g = (const float*)d_in[14]; const float* tn_b = (const float*)d_in[15];
    const float* ffn_w1 = (const float*)d_in[16]; const float* ffn_b1 = (const float*)d_in[17];
    const float* ffn_w2 = (const float*)d_in[18]; const float* ffn_b2 = (const float*)d_in[19];
    const float* fn_g = (const float*)d_in[20];  const float* fn_b = (const float*)d_in[21];
    const float* mfc_w = (const float*)d_in[22];
    const float* mln_g = (const float*)d_in[23]; const float* mln_b = (const float*)d_in[24];
    const float* fm_w = (const float*)d_in[25];  const float* fm_b = (const float*)d_in[26];

    // --- workspace layout (256B-aligned slices) ---
    char* wp = (char*)d_ws;
    auto alloc = [&](size_t bytes) -> char* {
        char* p = wp;
        wp += (bytes + 255) & ~(size_t)255;
        return p;
    };
    const int BN = B_ * N_;
    float* y     = (float*)alloc((size_t)B_ * C_ * HW_ * 4);       // 33.5 MB
    float* stats = (float*)alloc((size_t)B_ * G_ * 2 * 4);
    int*   tinds = (int*)  alloc((size_t)BN * TOPK * 4);
    float* tfeat = (float*)alloc((size_t)BN * TOPK * C_ * 4);      // 20.5 MB (reused as lowres)
    float* kfeat = (float*)alloc((size_t)BN * TOPK * C_ * 4);
    float* vfeat = (float*)alloc((size_t)BN * TOPK * C_ * 4);
    float* qbuf  = (float*)alloc((size_t)BN * C_ * 4);
    float* obuf  = (float*)alloc((size_t)BN * C_ * 4);
    float* obj1  = (float*)alloc((size_t)BN * C_ * 4);
    float* obj1n = (float*)alloc((size_t)BN * C_ * 4);
    float* ffnh  = (float*)alloc((size_t)BN * FFN_ * 4);
    float* obj2  = (float*)alloc((size_t)BN * C_ * 4);
    float* objf  = (float*)alloc((size_t)BN * C_ * 4);
    float* mtmp  = (float*)alloc((size_t)BN * C_ * 4);
    float* mbuf  = (float*)alloc((size_t)BN * C_ * 4);
    float* mfeat = (float*)alloc((size_t)BN * C_ * 4);

    float* out_mask = (float*)d_out;
    float* out_obj  = out_mask + (size_t)B_ * N_ * 4 * HW_;

    const int STRIP = 16 * MT;   // 64 rows per wave
    #define MTILES(M) ((unsigned)(((M) + STRIP - 1) / STRIP))

    // 1. 1x1 conv as batched GEMM: y[b] = conv_w (256x256) * x[b] (256xHW)
    wmma_gemm_f32<<<dim3(HW_ / 16, MTILES(C_), B_), 32, 0, stream>>>(
        conv_w, 0, C_, x, (long)C_ * HW_, HW_,
        nullptr, nullptr, 0, y, (long)C_ * HW_, HW_, C_, C_, 0);

    // 2-3. GroupNorm + ReLU
    gn_stats_kernel<<<B_ * G_, 256, 0, stream>>>(y, stats);
    gn_apply_kernel<<<(B_ * C_ * HW_) / 256, 256, 0, stream>>>(y, stats, gn_g, gn_b);

    // 4-5. top-K + gather
    topk_kernel<<<BN, 256, 0, stream>>>(preds, tinds);
    gather_kernel<<<BN * TOPK, 256, 0, stream>>>(y, tinds, tfeat);

    // 6-8. q/k/v projections
    wmma_gemm_f32<<<dim3(C_ / 16, MTILES(BN), 1), 32, 0, stream>>>(
        pfeat, 0, C_, wq, 0, C_, bq, nullptr, 0, qbuf, 0, C_, BN, C_, 0);
    wmma_gemm_f32<<<dim3(C_ / 16, MTILES(BN * TOPK), 1), 32, 0, stream>>>(
        tfeat, 0, C_, wk, 0, C_, bk, nullptr, 0, kfeat, 0, C_, BN * TOPK, C_, 0);
    wmma_gemm_f32<<<dim3(C_ / 16, MTILES(BN * TOPK), 1), 32, 0, stream>>>(
        tfeat, 0, C_, wv, 0, C_, bv, nullptr, 0, vfeat, 0, C_, BN * TOPK, C_, 0);

    // 9. attention core
    attn_kernel<<<BN, NH_ * 32, 0, stream>>>(qbuf, kfeat, vfeat, obuf);

    // 10-11. output proj + residual + LN(tn)
    wmma_gemm_f32<<<dim3(C_ / 16, MTILES(BN), 1), 32, 0, stream>>>(
        obuf, 0, C_, wo, 0, C_, bo, pfeat, 0, obj1, 0, C_, BN, C_, 0);
    layernorm_kernel<<<BN, C_, 0, stream>>>(obj1, tn_g, tn_b, obj1n, nullptr, 0);

    // 12-14. FFN + residual + LN(fn); also emit obj to d_out tail
    wmma_gemm_f32<<<dim3(FFN_ / 16, MTILES(BN), 1), 32, 0, stream>>>(
        obj1n, 0, C_, ffn_w1, 0, FFN_, ffn_b1, nullptr, 0, ffnh, 0, FFN_, BN, C_, FLAG_RELU);
    wmma_gemm_f32<<<dim3(C_ / 16, MTILES(BN), 1), 32, 0, stream>>>(
        ffnh, 0, FFN_, ffn_w2, 0, C_, ffn_b2, obj1n, 0, obj2, 0, C_, BN, FFN_, 0);
    layernorm_kernel<<<BN, C_, 0, stream>>>(obj2, fn_g, fn_b, objf, out_obj, 0);

    // 15. three mask-FC layers: m = relu(LN(m @ mfc_w[i]))
    const float* mi = objf;
    for (int i = 0; i < 3; ++i) {
        wmma_gemm_f32<<<dim3(C_ / 16, MTILES(BN), 1), 32, 0, stream>>>(
            mi, 0, C_, mfc_w + (size_t)i * C_ * C_, 0, C_,
            nullptr, nullptr, 0, mtmp, 0, C_, BN, C_, 0);
        layernorm_kernel<<<BN, C_, 0, stream>>>(mtmp, mln_g + i * C_, mln_b + i * C_,
                                                mbuf, nullptr, 1);
        mi = mbuf;
    }

    // 16. mask_feat = m @ fm_w + fm_b
    wmma_gemm_f32<<<dim3(C_ / 16, MTILES(BN), 1), 32, 0, stream>>>(
        mbuf, 0, C_, fm_w, 0, C_, fm_b, nullptr, 0, mfeat, 0, C_, BN, C_, 0);

    // 17. final einsum per batch: lowres[b] = mask_feat[b] (100x256) * y[b] (256xHW)
    float* lowres = tfeat;  // tfeat no longer needed
    wmma_gemm_f32<<<dim3(HW_ / 16, MTILES(N_), B_), 32, 0, stream>>>(
        mfeat, (long)N_ * C_, C_, y, (long)C_ * HW_, HW_,
        nullptr, nullptr, 0, lowres, (long)N_ * HW_, HW_, N_, C_, 0);

    // 18. bilinear 2x upsample -> d_out
    const long tot = (long)B_ * N_ * 4 * HW_;
    upsample_kernel<<<(unsigned)((tot + 255) / 256), 256, 0, stream>>>(lowres, out_mask);
}